// deepedgeNet_14224931685026
// MI455X (gfx1250) — compile-verified
//
#include <hip/hip_runtime.h>
#include <cstddef>

typedef __attribute__((ext_vector_type(2))) float v2f;
typedef __attribute__((ext_vector_type(8))) float v8f;

// ---------------------------------------------------------------------------
// utility: zero a float buffer
// ---------------------------------------------------------------------------
__global__ void zero_f32(float* __restrict__ p, size_t n) {
    size_t i = (size_t)blockIdx.x * blockDim.x + threadIdx.x;
    if (i < n) p[i] = 0.0f;
}

// ---------------------------------------------------------------------------
// degree count: deg[dst[e]] += 1   (fp32 L2 atomics)
// ---------------------------------------------------------------------------
__global__ void degree_count(const int* __restrict__ dst, float* __restrict__ deg, int E) {
    int e = blockIdx.x * blockDim.x + threadIdx.x;
    if (e < E) unsafeAtomicAdd(&deg[dst[e]], 1.0f);
}

__global__ void recip_clamp(float* __restrict__ deg, int n) {
    int i = blockIdx.x * blockDim.x + threadIdx.x;
    if (i < n) deg[i] = 1.0f / fmaxf(deg[i], 1.0f);
}

// ---------------------------------------------------------------------------
// edge scatter, din == 64: 16 threads per edge, float4 gather + 4 fp32 atomics
// ---------------------------------------------------------------------------
__global__ void scatter64(const float* __restrict__ x, const int* __restrict__ src,
                          const int* __restrict__ dst, float* __restrict__ agg, int E) {
    int tid = blockIdx.x * blockDim.x + threadIdx.x;
    int e = tid >> 4;
    if (e >= E) return;
    int f = (tid & 15) << 2;
    int s = src[e];
    int d = dst[e];
    const float4 v = *reinterpret_cast<const float4*>(x + (size_t)s * 64 + f);
    float* p = agg + (size_t)d * 64 + f;
    unsafeAtomicAdd(p + 0, v.x);
    unsafeAtomicAdd(p + 1, v.y);
    unsafeAtomicAdd(p + 2, v.z);
    unsafeAtomicAdd(p + 3, v.w);
}

// edge scatter, din == 3 (input features)
__global__ void scatter3(const float* __restrict__ x, const int* __restrict__ src,
                         const int* __restrict__ dst, float* __restrict__ agg, int E) {
    int e = blockIdx.x * blockDim.x + threadIdx.x;
    if (e >= E) return;
    int s = src[e];
    int d = dst[e];
    float v0 = x[(size_t)s * 3 + 0];
    float v1 = x[(size_t)s * 3 + 1];
    float v2 = x[(size_t)s * 3 + 2];
    float* p = agg + (size_t)d * 3;
    unsafeAtomicAdd(p + 0, v0);
    unsafeAtomicAdd(p + 1, v1);
    unsafeAtomicAdd(p + 2, v2);
}

// ---------------------------------------------------------------------------
// Fused layer: Y = act( X@Ws + (AGG * deg_inv[row]) @ Wn + b )
// 128 threads = 4 wave32 per block; each wave computes 16 rows x DOUT cols
// using V_WMMA_F32_16X16X4_F32 tiles (full fp32 precision).
//
// Weights are stored TRANSPOSED and zero-PADDED in LDS (sW[n][k]), so the
// inner loop has no guards: B fragments are ds_load_2addr_b64 pairs, A
// fragments are unconditional global loads with row clamped and validity
// folded into a multiplicative mask (selects, never EXEC divergence ->
// WMMA always runs with EXEC all-ones).
//
// Epilogue: wave-uniform full-tile test (readfirstlane-forced scalar branch)
// selects an unguarded store loop; only the single partial block takes the
// per-row guarded path.
//
// ISA 7.12.2 f32 layouts (wave32):
//   A 16x4 : lane m=l&15 is row; VGPR0/1 hold K = (l>>4)*2 + {0,1}
//   B 4x16 : lane n=l&15 is col; VGPR0/1 hold K = (l>>4)*2 + {0,1}
//   C/D    : VGPR j -> row (l>>4)*8 + j, col l&15
// ---------------------------------------------------------------------------
template <int DIN, int DOUT, bool RELU>
__global__ __launch_bounds__(128) void gcn_layer(
    const float* __restrict__ X, const float* __restrict__ AGG,
    const float* __restrict__ deg_inv,
    const float* __restrict__ Ws, const float* __restrict__ Wn,
    const float* __restrict__ bias, float* __restrict__ Y, int n_rows) {
    constexpr int NT   = (DOUT + 15) / 16;  // 16-col output tiles
    constexpr int KT   = (DIN + 3) / 4;     // K steps of 4
    constexpr int KPAD = KT * 4;
    constexpr int NPAD = NT * 16;

    __shared__ float sW[2][NPAD * KPAD];  // transposed [n*KPAD + k]; 0=Ws, 1=Wn

    // zero-fill padding, then scatter-transpose the weights into LDS
    for (int i = threadIdx.x; i < NPAD * KPAD; i += 128) {
        sW[0][i] = 0.0f;
        sW[1][i] = 0.0f;
    }
    __syncthreads();
    for (int i = threadIdx.x; i < DIN * DOUT; i += 128) {
        const int k = i / DOUT;
        const int n = i % DOUT;
        sW[0][n * KPAD + k] = Ws[i];
        sW[1][n * KPAD + k] = Wn[i];
    }
    __syncthreads();

    const int wave = threadIdx.x >> 5;  // wave32
    const int lane = threadIdx.x & 31;
    const int half = lane >> 4;  // 0 or 1
    const int m    = lane & 15;
    const int kb   = half * 2;
    const int row0 = blockIdx.x * 64 + wave * 16;
    const int r    = row0 + m;
    const int rr   = r < n_rows ? r : (n_rows - 1);  // clamped row (safe load)
    const float vmask = r < n_rows ? 1.0f : 0.0f;    // validity as a multiply

    v8f acc[NT] = {};

    #pragma unroll
    for (int pass = 0; pass < 2; ++pass) {
        const float* __restrict__ A = pass ? AGG : X;
        const float* __restrict__ W = pass ? sW[1] : sW[0];
        const float scale = (pass ? deg_inv[rr] : 1.0f) * vmask;
        #pragma unroll
        for (int kt = 0; kt < KT; ++kt) {
            const int k0 = kt * 4 + kb;  // this lane's K pair
            v2f a;
            if constexpr (DIN % 4 == 0) {
                // contiguous, 8B-aligned: one global_load_b64
                const float2 t = *reinterpret_cast<const float2*>(&A[(size_t)rr * DIN + k0]);
                a.x = t.x * scale;
                a.y = t.y * scale;
            } else {
                // unconditional clamped loads + selects (no EXEC divergence)
                const int k0c = k0 < DIN - 1 ? k0 : (DIN - 1);
                const int k1c = k0 + 1 < DIN - 1 ? k0 + 1 : (DIN - 1);
                const float t0 = A[(size_t)rr * DIN + k0c];
                const float t1 = A[(size_t)rr * DIN + k1c];
                a.x = (k0 < DIN) ? t0 * scale : 0.0f;
                a.y = (k0 + 1 < DIN) ? t1 * scale : 0.0f;
            }
            #pragma unroll
            for (int nt = 0; nt < NT; ++nt) {
                // B fragment: one ds_load_b64 from the transposed padded tile
                const v2f b = *reinterpret_cast<const v2f*>(&W[(nt * 16 + m) * KPAD + k0]);
                acc[nt] = __builtin_amdgcn_wmma_f32_16x16x4_f32(
                    false, a, false, b, (short)0, acc[nt], false, false);
            }
        }
    }

    // epilogue: bias + leaky relu + store
    const int mr0 = row0 + half * 8;
    // row0 is wave-uniform; force scalar so the tile-full test is one s_cbranch
    const bool full_tile = (__builtin_amdgcn_readfirstlane(row0) + 16) <= n_rows;

    if (full_tile) {
        #pragma unroll
        for (int nt = 0; nt < NT; ++nt) {
            const int c = nt * 16 + m;
            if (c < DOUT) {  // statically true for DOUT=64
                const float bc = bias[c];
                #pragma unroll
                for (int j = 0; j < 8; ++j) {
                    float v = acc[nt][j] + bc;
                    if (RELU) v = v > 0.0f ? v : 0.01f * v;
                    Y[(size_t)(mr0 + j) * DOUT + c] = v;
                }
            }
        }
    } else {
        #pragma unroll
        for (int nt = 0; nt < NT; ++nt) {
            const int c = nt * 16 + m;
            if (c < DOUT) {
                const float bc = bias[c];
                #pragma unroll
                for (int j = 0; j < 8; ++j) {
                    const int gr = mr0 + j;
                    if (gr < n_rows) {
                        float v = acc[nt][j] + bc;
                        if (RELU) v = v > 0.0f ? v : 0.01f * v;
                        Y[(size_t)gr * DOUT + c] = v;
                    }
                }
            }
        }
    }
}

// ---------------------------------------------------------------------------
// launch
// ---------------------------------------------------------------------------
extern "C" void kernel_launch(void* const* d_in, const int* in_sizes, int n_in,
                              void* d_out, int out_size, void* d_ws, size_t ws_size,
                              hipStream_t stream) {
    const float* feats = (const float*)d_in[0];
    const int* src = (const int*)d_in[1];
    const int* dst = (const int*)d_in[2];
    const float* Ws[5];
    const float* Wn[5];
    const float* bb[5];
    for (int i = 0; i < 5; ++i) {
        Ws[i] = (const float*)d_in[3 + 3 * i];
        Wn[i] = (const float*)d_in[4 + 3 * i];
        bb[i] = (const float*)d_in[5 + 3 * i];
    }
    const int N = in_sizes[0] / 3;
    const int E = in_sizes[1];

    float* wsf  = (float*)d_ws;
    float* xa   = wsf;                       // N*64
    float* xb   = xa + (size_t)N * 64;       // N*64
    float* agg  = xb + (size_t)N * 64;       // N*64
    float* dinv = agg + (size_t)N * 64;      // N

    auto zero = [&](float* p, size_t n) {
        zero_f32<<<(unsigned)((n + 255) / 256), 256, 0, stream>>>(p, n);
    };

    // degree -> 1/max(deg,1)
    zero(dinv, (size_t)N);
    degree_count<<<(E + 255) / 256, 256, 0, stream>>>(dst, dinv, E);
    recip_clamp<<<(N + 255) / 256, 256, 0, stream>>>(dinv, N);

    const int ggrid = (N + 63) / 64;
    const int sgrid64 = (int)(((size_t)E * 16 + 255) / 256);

    // layer 1: din=3 -> 64, lrelu
    zero(agg, (size_t)N * 3);
    scatter3<<<(E + 255) / 256, 256, 0, stream>>>(feats, src, dst, agg, E);
    gcn_layer<3, 64, true><<<ggrid, 128, 0, stream>>>(feats, agg, dinv, Ws[0], Wn[0], bb[0], xa, N);

    // layer 2: 64 -> 64, lrelu
    zero(agg, (size_t)N * 64);
    scatter64<<<sgrid64, 256, 0, stream>>>(xa, src, dst, agg, E);
    gcn_layer<64, 64, true><<<ggrid, 128, 0, stream>>>(xa, agg, dinv, Ws[1], Wn[1], bb[1], xb, N);

    // layer 3
    zero(agg, (size_t)N * 64);
    scatter64<<<sgrid64, 256, 0, stream>>>(xb, src, dst, agg, E);
    gcn_layer<64, 64, true><<<ggrid, 128, 0, stream>>>(xb, agg, dinv, Ws[2], Wn[2], bb[2], xa, N);

    // layer 4
    zero(agg, (size_t)N * 64);
    scatter64<<<sgrid64, 256, 0, stream>>>(xa, src, dst, agg, E);
    gcn_layer<64, 64, true><<<ggrid, 128, 0, stream>>>(xa, agg, dinv, Ws[3], Wn[3], bb[3], xb, N);

    // layer 5: 64 -> 3, no activation, straight to d_out
    zero(agg, (size_t)N * 64);
    scatter64<<<sgrid64, 256, 0, stream>>>(xb, src, dst, agg, E);
    gcn_layer<64, 3, false><<<ggrid, 128, 0, stream>>>(xb, agg, dinv, Ws[4], Wn[4], bb[4],
                                                       (float*)d_out, N);
}